// VNFrameEstimator_86681029968414
// MI455X (gfx1250) — compile-verified
//
#include <hip/hip_runtime.h>
#include <math.h>

typedef float v2f __attribute__((ext_vector_type(2)));
typedef float v8f __attribute__((ext_vector_type(8)));

#define WAVES_PER_BLOCK 8

struct F3 { float x, y, z; };

__device__ __forceinline__ F3 f3(float x, float y, float z){ F3 r{x,y,z}; return r; }
__device__ __forceinline__ F3 cross3(F3 a, F3 b){
  return f3(a.y*b.z - a.z*b.y, a.z*b.x - a.x*b.z, a.x*b.y - a.y*b.x);
}
__device__ __forceinline__ float dot3(F3 a, F3 b){ return a.x*b.x + a.y*b.y + a.z*b.z; }
__device__ __forceinline__ F3 scale3(F3 a, float s){ return f3(a.x*s, a.y*s, a.z*s); }
__device__ __forceinline__ F3 sub3(F3 a, F3 b){ return f3(a.x-b.x, a.y-b.y, a.z-b.z); }

// Eigenvector of symmetric 3x3 for eigenvalue lam: largest cross product of
// rows of (M - lam*I), normalized.
__device__ F3 sym_eigvec(float m00, float m01, float m02,
                         float m11, float m12, float m22, float lam)
{
  F3 r0 = f3(m00 - lam, m01, m02);
  F3 r1 = f3(m01, m11 - lam, m12);
  F3 r2 = f3(m02, m12, m22 - lam);
  F3 c0 = cross3(r0, r1), c1 = cross3(r0, r2), c2 = cross3(r1, r2);
  float d0 = dot3(c0,c0), d1 = dot3(c1,c1), d2 = dot3(c2,c2);
  F3 best = c0; float db = d0;
  if (d1 > db){ best = c1; db = d1; }
  if (d2 > db){ best = c2; db = d2; }
  if (db < 1e-38f) return f3(1.f, 0.f, 0.f);
  return scale3(best, rsqrtf(db));
}

__global__ __launch_bounds__(256)
void vnframe_kernel(const float* __restrict__ vfeat, float* __restrict__ out, int nB)
{
  __shared__ float vnsh[WAVES_PER_BLOCK][128*4];   // per-wave: 128 rows x (xyz + 0-pad)
  const int lane = threadIdx.x & 31;
  const int wave = threadIdx.x >> 5;
  long b = (long)blockIdx.x * WAVES_PER_BLOCK + wave;
  if (b >= nB) b = nB - 1;   // tail waves duplicate last batch (same data rewritten)

  // ---- stream 12 floats (4 channels) per lane, fully coalesced b128 loads ----
  const float4* src = (const float4*)vfeat + b*96 + lane*3;
  float4 ld0 = src[0], ld1 = src[1], ld2 = src[2];
  float cx[4] = {ld0.x, ld0.w, ld1.z, ld2.y};
  float cy[4] = {ld0.y, ld1.x, ld1.w, ld2.z};
  float cz[4] = {ld0.z, ld1.y, ld2.x, ld2.w};

  float sx = 0.f, sy = 0.f, sz = 0.f;
  float4* vrow = (float4*)&vnsh[wave][0];
#pragma unroll
  for (int t = 0; t < 4; ++t) {
    float d   = cx[t]*cx[t] + cy[t]*cy[t] + cz[t]*cz[t];
    float den = fmaxf(sqrtf(d), 1e-12f);            // EPS from reference
    float inv = 1.0f / den;
    float nx = cx[t]*inv, ny = cy[t]*inv, nz = cz[t]*inv;
    sx += nx; sy += ny; sz += nz;
    vrow[lane*4 + t] = make_float4(nx, ny, nz, 0.0f);
  }
  __syncthreads();

  // ---- wave reduction of sum(v) for the sign fix ----
#pragma unroll
  for (int m = 16; m > 0; m >>= 1) {
    sx += __shfl_xor(sx, m, 32);
    sy += __shfl_xor(sy, m, 32);
    sz += __shfl_xor(sz, m, 32);
  }
  F3 sumv = f3(sx, sy, sz);

  float m00, m01, m02, m11, m12, m22;

#if __has_builtin(__builtin_amdgcn_wmma_f32_16x16x4_f32)
  // Gram matrix M = V^T V via V_WMMA_F32_16X16X4_F32, K=4 channels/issue.
  // Symmetric fill => A and B operand registers are identical:
  //   lanes 0-15 : {v[c0+0][i], v[c0+1][i]},  lanes 16-31 : {v[c0+2][i], v[c0+3][i]}
  // Only D[0..2][0..2] is read, so unused rows/cols may hold anything
  // (pad slot supplies 0 for i>=3 anyway).
  {
    const float* bw = &vnsh[wave][0];
    const int i  = lane & 15;
    const int ic = (i < 3) ? i : 3;            // clamp to zero pad slot
    const int half = (lane >> 4) << 1;         // 0 for lanes 0-15, 2 for 16-31
    v8f acc = {0.f,0.f,0.f,0.f,0.f,0.f,0.f,0.f};
#pragma unroll
    for (int ch = 0; ch < 32; ++ch) {
      const int row = ch*4 + half;
      v2f a;
      a.x = bw[row*4 + ic];
      a.y = bw[row*4 + 4 + ic];
      acc = __builtin_amdgcn_wmma_f32_16x16x4_f32(false, a, false, a,
                                                  (short)0, acc, false, false);
    }
    // D layout: VGPR r, lane n (n<16) holds M[r][n]
    m00 = __shfl(acc[0], 0, 32);
    m01 = __shfl(acc[0], 1, 32);
    m02 = __shfl(acc[0], 2, 32);
    m11 = __shfl(acc[1], 1, 32);
    m12 = __shfl(acc[1], 2, 32);
    m22 = __shfl(acc[2], 2, 32);
  }
#else
  { // VALU fallback (keeps compile green if builtin name differs)
    const float* bw = &vnsh[wave][0];
    float a00=0,a01=0,a02=0,a11=0,a12=0,a22=0;
    for (int c = lane; c < 128; c += 32) {
      float x = bw[c*4+0], y = bw[c*4+1], z = bw[c*4+2];
      a00 += x*x; a01 += x*y; a02 += x*z; a11 += y*y; a12 += y*z; a22 += z*z;
    }
#pragma unroll
    for (int m = 16; m > 0; m >>= 1) {
      a00 += __shfl_xor(a00, m, 32); a01 += __shfl_xor(a01, m, 32);
      a02 += __shfl_xor(a02, m, 32); a11 += __shfl_xor(a11, m, 32);
      a12 += __shfl_xor(a12, m, 32); a22 += __shfl_xor(a22, m, 32);
    }
    m00=a00; m01=a01; m02=a02; m11=a11; m12=a12; m22=a22;
  }
#endif

  // diag noise
  m00 += 1e-5f; m11 += 2e-5f; m22 += 3e-5f;

  // ---- analytic eigenvalues (trigonometric method), ascending order ----
  float q  = (m00 + m11 + m22) * (1.0f/3.0f);
  float pp1 = m01*m01 + m02*m02 + m12*m12;
  float a00 = m00 - q, a11 = m11 - q, a22 = m22 - q;
  float pp2 = a00*a00 + a11*a11 + a22*a22 + 2.0f*pp1;
  float p  = sqrtf(pp2 * (1.0f/6.0f));
  float lam_min, lam_mid;
  if (p < 1e-30f) {
    lam_min = q; lam_mid = q;
  } else {
    float ip  = 1.0f / p;
    float b00 = a00*ip, b11 = a11*ip, b22 = a22*ip;
    float b01 = m01*ip, b02 = m02*ip, b12 = m12*ip;
    float detB = b00*(b11*b22 - b12*b12)
               - b01*(b01*b22 - b12*b02)
               + b02*(b01*b12 - b11*b02);
    float r = fminf(1.0f, fmaxf(-1.0f, 0.5f * detB));
    float phi = acosf(r) * (1.0f/3.0f);
    float lam_max = q + 2.0f*p*cosf(phi);
    lam_min = q + 2.0f*p*cosf(phi + 2.0943951023931953f); // + 2*pi/3
    lam_mid = 3.0f*q - lam_max - lam_min;
  }

  F3 v1 = sym_eigvec(m00,m01,m02,m11,m12,m22, lam_min);
  F3 v2 = sym_eigvec(m00,m01,m02,m11,m12,m22, lam_mid);

  // Gram-Schmidt v2 against v1 (robust near-degenerate case)
  v2 = sub3(v2, scale3(v1, dot3(v2, v1)));
  float l2 = dot3(v2, v2);
  if (l2 > 1e-24f) {
    v2 = scale3(v2, rsqrtf(l2));
  } else {
    F3 h = (fabsf(v1.x) < 0.9f) ? f3(1.f,0.f,0.f) : f3(0.f,1.f,0.f);
    F3 w = cross3(v1, h);
    v2 = scale3(w, rsqrtf(fmaxf(dot3(w,w), 1e-30f)));
  }

  // sign fix: sign(dot(sum v, vec)), sign(0) -> +1
  float s1 = (dot3(sumv, v1) < 0.0f) ? -1.0f : 1.0f;
  float s2 = (dot3(sumv, v2) < 0.0f) ? -1.0f : 1.0f;
  v1 = scale3(v1, s1);
  v2 = scale3(v2, s2);
  F3 v3 = cross3(v1, v2);

  // out[b][i][j], j: 0=v1, 1=v2, 2=v3 ; lanes 0..8 store one float each
  float ov = (lane==0) ? v1.x : (lane==1) ? v2.x : (lane==2) ? v3.x :
             (lane==3) ? v1.y : (lane==4) ? v2.y : (lane==5) ? v3.y :
             (lane==6) ? v1.z : (lane==7) ? v2.z : v3.z;
  if (lane < 9) out[b*9 + lane] = ov;
}

extern "C" void kernel_launch(void* const* d_in, const int* in_sizes, int n_in,
                              void* d_out, int out_size, void* d_ws, size_t ws_size,
                              hipStream_t stream) {
  const float* vfeat = (const float*)d_in[0];
  float* out = (float*)d_out;
  const int nB = in_sizes[0] / 384;                 // (B,128,3) -> B
  const int blocks = (nB + WAVES_PER_BLOCK - 1) / WAVES_PER_BLOCK;
  vnframe_kernel<<<blocks, 256, 0, stream>>>(vfeat, out, nB);
}